// Representation_21801253994878
// MI455X (gfx1250) — compile-verified
//
#include <hip/hip_runtime.h>

// ---------------- problem constants ----------------
#define BATCH   16
#define NNODE   1000
#define NTOT    (BATCH * NNODE)          // 16000
#define NFEAT   16
#define EDGE_B  15984                    // MAXN*(N_NODE-1)
#define ETOT    (BATCH * EDGE_B)         // 255744
#define RW      (NNODE*NFEAT + 2*EDGE_B) // 47968  (row width of r)
#define SLEFT   (NNODE*NFEAT)            // 16000
#define OW      (NNODE*64 + 2*EDGE_B)    // 95968  (row width of output)
#define NEG_SLOPE 0.2f
#define LN_EPS    1e-5f

typedef float v2f __attribute__((ext_vector_type(2)));
typedef float v8f __attribute__((ext_vector_type(8)));

// order-preserving encoding of f32 into u32 so atomicMax(u32) == float max
__device__ __forceinline__ unsigned fenc(float f) {
    unsigned u = __float_as_uint(f);
    return (u & 0x80000000u) ? ~u : (u | 0x80000000u);
}
__device__ __forceinline__ float fdec(unsigned u) {
    return __uint_as_float((u & 0x80000000u) ? (u ^ 0x80000000u) : ~u);
}
#define ENC_NEG_INF 0x007FFFFFu   // fenc(-inf)

// ---------------- 0) parse r: node feats + edge indices ----------------
__global__ __launch_bounds__(256) void parse_k(const float* __restrict__ r,
                                               float* __restrict__ X0,
                                               int* __restrict__ snd,
                                               int* __restrict__ rcv) {
    int i = blockIdx.x * 256 + threadIdx.x;
    if (i < BATCH * SLEFT) {                       // contiguous per-batch feature copy
        int b = i / SLEFT, t = i - b * SLEFT;
        X0[i] = r[b * RW + t];
    }
    if (i < ETOT) {                                // float -> int edge ids + batch offset
        int b = i / EDGE_B, j = i - b * EDGE_B;
        snd[i] = (int)r[b * RW + SLEFT + j]          + b * NNODE;
        rcv[i] = (int)r[b * RW + SLEFT + EDGE_B + j] + b * NNODE;
    }
}

// ---------------- 1) fp32 GEMM via V_WMMA_F32_16X16X4_F32 ----------------
// One wave (32 lanes) computes one 16x16 tile of H = X[M,K] @ W[K,N].
__global__ __launch_bounds__(32) void gemm_wmma_k(const float* __restrict__ X,
                                                  const float* __restrict__ W,
                                                  float* __restrict__ H,
                                                  int K, int N) {
    const int lane = threadIdx.x;
    const int tm = blockIdx.x * 16;
    const int tn = blockIdx.y * 16;
    const int hi = lane >> 4;       // 0: lanes 0-15, 1: lanes 16-31
    const int lo = lane & 15;

    v8f acc = {};
    for (int k0 = 0; k0 < K; k0 += 4) {
        // A 16x4 f32: lane m=lo holds K = k0+2*hi (+0,+1) in VGPR0/1
        const float* xr = X + (tm + lo) * K + k0 + 2 * hi;
        v2f a; a.x = xr[0]; a.y = xr[1];
        // B 4x16 f32: lane n=lo, rows K = k0+2*hi (+0,+1)
        const float* wr = W + (k0 + 2 * hi) * N + tn + lo;
        v2f b; b.x = wr[0]; b.y = wr[N];
        acc = __builtin_amdgcn_wmma_f32_16x16x4_f32(
            /*neg_a=*/false, a, /*neg_b=*/false, b,
            /*c_mod=*/(short)0, acc, /*reuse_a=*/false, /*reuse_b=*/false);
    }
    // C/D layout: VGPR i -> row tm + i + 8*hi, col tn + lo
    float* hr = H + (tm + 8 * hi) * N + tn + lo;
#pragma unroll
    for (int i = 0; i < 8; ++i) hr[i * N] = acc[i];
}

// ---------------- 2) per-node attention scores (wave reduction) ----------------
__global__ __launch_bounds__(256) void scores_k(const float* __restrict__ H,
                                                const float* __restrict__ asrc,
                                                const float* __restrict__ adst,
                                                float* __restrict__ es,
                                                float* __restrict__ ed, int N) {
    int node = blockIdx.x * 8 + (threadIdx.x >> 5);
    int lane = threadIdx.x & 31;
    if (node >= NTOT) return;
    const float* h = H + (size_t)node * N;
    float s = 0.f, d = 0.f;
    for (int c = lane; c < N; c += 32) { float v = h[c]; s += v * asrc[c]; d += v * adst[c]; }
#pragma unroll
    for (int o = 16; o; o >>= 1) { s += __shfl_xor(s, o, 32); d += __shfl_xor(d, o, 32); }
    if (lane == 0) { es[node] = s; ed[node] = d; }
}

// ---------------- 3) init accumulators ----------------
__global__ __launch_bounds__(256) void init_k(float* __restrict__ O, unsigned* __restrict__ m,
                                              float* __restrict__ denom, int nO) {
    int i = blockIdx.x * 256 + threadIdx.x;
    if (i < nO) O[i] = 0.f;
    if (i < NTOT) { m[i] = ENC_NEG_INF; denom[i] = 0.f; }
}

// ---------------- 4) edge pass 1: e = leaky_relu(es[s]+ed[r]); segment max ----------------
__global__ __launch_bounds__(256) void edge_e_k(const int* __restrict__ snd,
                                                const int* __restrict__ rcv,
                                                const float* __restrict__ es,
                                                const float* __restrict__ ed,
                                                float* __restrict__ e,
                                                unsigned* __restrict__ m) {
    int i = blockIdx.x * 256 + threadIdx.x;
    if (i >= ETOT) return;
    float v = es[snd[i]] + ed[rcv[i]];
    v = (v > 0.f) ? v : NEG_SLOPE * v;
    e[i] = v;
    atomicMax(&m[rcv[i]], fenc(v));
}

// ---------------- 5) edge pass 2: ex = exp(e - m[r]); segment sum ----------------
__global__ __launch_bounds__(256) void edge_ex_k(const int* __restrict__ rcv,
                                                 float* __restrict__ e,
                                                 const unsigned* __restrict__ m,
                                                 float* __restrict__ denom) {
    int i = blockIdx.x * 256 + threadIdx.x;
    if (i >= ETOT) return;
    int rr = rcv[i];
    float ex = __expf(e[i] - fdec(m[rr]));
    e[i] = ex;
    atomicAdd(&denom[rr], ex);
}

// ---------------- 6) aggregate: O[r] += (ex/denom[r]) * H[s] ----------------
__global__ void agg_k(const int* __restrict__ snd, const int* __restrict__ rcv,
                      const float* __restrict__ e, const float* __restrict__ denom,
                      const float* __restrict__ H, float* __restrict__ O, int N) {
    int i = blockIdx.x;           // edge
    int c = threadIdx.x;          // channel (blockDim.x == N)
    int rr = rcv[i];
    float coef = e[i] / denom[rr];
    atomicAdd(&O[(size_t)rr * N + c], coef * H[(size_t)snd[i] * N + c]);
}

// ---------------- 7) LayerNorm over node axis (per b,c) + ReLU, in place ----------------
__global__ __launch_bounds__(256) void ln_relu_k(float* __restrict__ X,
                                                 const float* __restrict__ sc,
                                                 const float* __restrict__ of, int C) {
    int c = blockIdx.x, b = blockIdx.y, t = threadIdx.x;
    __shared__ float ssum[256], ssq[256];
    float s = 0.f, q = 0.f;
    for (int n = t; n < NNODE; n += 256) {
        float v = X[((size_t)(b * NNODE + n)) * C + c];
        s += v; q += v * v;
    }
    ssum[t] = s; ssq[t] = q; __syncthreads();
    for (int o = 128; o; o >>= 1) {
        if (t < o) { ssum[t] += ssum[t + o]; ssq[t] += ssq[t + o]; }
        __syncthreads();
    }
    float mean = ssum[0] * (1.f / NNODE);
    float var  = ssq[0] * (1.f / NNODE) - mean * mean;
    float rs   = rsqrtf(var + LN_EPS) * sc[c];
    float off  = of[c];
    for (int n = t; n < NNODE; n += 256) {
        size_t idx = ((size_t)(b * NNODE + n)) * C + c;
        float y = (X[idx] - mean) * rs + off;
        X[idx] = y > 0.f ? y : 0.f;
    }
}

// ---------------- 8) final write: s + b4, then pass-through edge floats ----------------
__global__ __launch_bounds__(256) void final_k(const float* __restrict__ X,
                                               const float* __restrict__ b4,
                                               const float* __restrict__ r,
                                               float* __restrict__ out) {
    int i = blockIdx.x * 256 + threadIdx.x;
    if (i >= BATCH * OW) return;
    int b = i / OW, j = i - b * OW;
    if (j < NNODE * 64) out[i] = X[(size_t)b * NNODE * 64 + j] + b4[j & 63];
    else                out[i] = r[(size_t)b * RW + SLEFT + (j - NNODE * 64)];
}

// ---------------- host orchestration ----------------
extern "C" void kernel_launch(void* const* d_in, const int* in_sizes, int n_in,
                              void* d_out, int out_size, void* d_ws, size_t ws_size,
                              hipStream_t stream) {
    (void)in_sizes; (void)n_in; (void)out_size; (void)ws_size;
    const float* r = (const float*)d_in[0];
    // dict order: W1,a_src1,a_dst1,b1, W2,..., b4, ln_s1,ln_o1,ln_s2,ln_o2,ln_s3,ln_o3
    const float* W[4]    = {(const float*)d_in[1], (const float*)d_in[5],
                            (const float*)d_in[9], (const float*)d_in[13]};
    const float* Asrc[4] = {(const float*)d_in[2], (const float*)d_in[6],
                            (const float*)d_in[10], (const float*)d_in[14]};
    const float* Adst[4] = {(const float*)d_in[3], (const float*)d_in[7],
                            (const float*)d_in[11], (const float*)d_in[15]};
    const float* b4      = (const float*)d_in[16];
    const float* LNs[3]  = {(const float*)d_in[17], (const float*)d_in[19], (const float*)d_in[21]};
    const float* LNo[3]  = {(const float*)d_in[18], (const float*)d_in[20], (const float*)d_in[22]};

    // workspace layout (float units)
    float*    ws    = (float*)d_ws;
    float*    xbuf  = ws;                         // 16000*256 (layer input / agg output)
    float*    hbuf  = xbuf + (size_t)NTOT * 256;  // 16000*256 (h = x@W)
    float*    ebuf  = hbuf + (size_t)NTOT * 256;  // 255744    (e then ex)
    int*      snd   = (int*)(ebuf + ETOT);        // 255744
    int*      rcv   = snd + ETOT;                 // 255744
    float*    es    = (float*)(rcv + ETOT);       // 16000
    float*    ed    = es + NTOT;                  // 16000
    unsigned* mbuf  = (unsigned*)(ed + NTOT);     // 16000
    float*    denom = (float*)(mbuf + NTOT);      // 16000

    const int Kdim[4] = {NFEAT, 256, 128, 128};
    const int Ndim[4] = {256, 128, 128, 64};

    // parse r
    {
        int nmax = BATCH * SLEFT;                 // 256000 > ETOT
        parse_k<<<(nmax + 255) / 256, 256, 0, stream>>>(r, xbuf, snd, rcv);
    }

    for (int L = 0; L < 4; ++L) {
        const int K = Kdim[L], N = Ndim[L];
        // h = x @ W  (WMMA fp32)
        gemm_wmma_k<<<dim3(NTOT / 16, N / 16), 32, 0, stream>>>(xbuf, W[L], hbuf, K, N);
        // per-node scores
        scores_k<<<(NTOT + 7) / 8, 256, 0, stream>>>(hbuf, Asrc[L], Adst[L], es, ed, N);
        // init accumulators (xbuf reused as aggregation output)
        {
            int nO = NTOT * N;
            init_k<<<(nO + 255) / 256, 256, 0, stream>>>(xbuf, mbuf, denom, nO);
        }
        // segmented softmax over edges
        edge_e_k<<<(ETOT + 255) / 256, 256, 0, stream>>>(snd, rcv, es, ed, ebuf, mbuf);
        edge_ex_k<<<(ETOT + 255) / 256, 256, 0, stream>>>(rcv, ebuf, mbuf, denom);
        // weighted message aggregation
        agg_k<<<ETOT, N, 0, stream>>>(snd, rcv, ebuf, denom, hbuf, xbuf, N);
        // LayerNorm(node axis) + ReLU for layers 0..2 (bias cancels under node-axis LN)
        if (L < 3)
            ln_relu_k<<<dim3(N, BATCH), 256, 0, stream>>>(xbuf, LNs[L], LNo[L], N);
    }

    // final output: s + b4 and edge-float pass-through
    final_k<<<(BATCH * OW + 255) / 256, 256, 0, stream>>>(xbuf, b4, r, (float*)d_out);
}